// Multimodel_Transformer_8203387536042
// MI455X (gfx1250) — compile-verified
//
#include <hip/hip_runtime.h>
#include <hip/hip_bf16.h>

// ---------------------------------------------------------------------------
// Multimodal graph-transformer forward for MI455X (gfx1250, wave32, WMMA).
//
// Input pointer layout assumption (compile-only environment, documented):
//   d_in[0] X0 (2,512,256) f32      d_in[1] X1        d_in[2] E (2,512,512,64)
//   d_in[3] y (2,128)               d_in[4] base_GRN  d_in[5] t_init (2,1)
//   d_in[6] node_mask (2,512) bool  d_in[7..] params flattened jax-tree order:
//     attn dict keys sorted: bg_add,bg_mul,e_add,e_mul,e_out,e_y,k,q,v,x_out,
//       x_y,y_e_add,y_e_mul,y_out1,y_out2,y_x_add,y_x_mul,y_y  (b then w each)
//     tl1 then tl2: linE1,linE2,linX1,linX2,lin_y1,lin_y2 (b,w) then
//       normE1,normE2,normX1,normX2,norm_y1,norm_y2 (b,g)
// Output: concat(X0o[262144], X1o[262144], E2[33554432], y2[256]) f32.
// ---------------------------------------------------------------------------

typedef float v8f __attribute__((ext_vector_type(8)));
typedef float v2f __attribute__((ext_vector_type(2)));
typedef int vi4 __attribute__((vector_size(4 * sizeof(int))));

#define BSZ 2
#define NN 512
#define DXX 256
#define DEE 64
#define DYY 128
#define NHH 8
#define DFF 32

#if defined(__has_builtin)
#if __has_builtin(__builtin_amdgcn_global_load_async_to_lds_b128) && \
    __has_builtin(__builtin_amdgcn_s_wait_asynccnt)
#define USE_ASYNC_LDS 1
#endif
#endif
#ifndef USE_ASYNC_LDS
#define USE_ASYNC_LDS 0
#endif

__device__ __forceinline__ v8f wmma4(v2f a, v2f b, v8f c) {
  // D = A(16x4,f32) * B(4x16,f32) + C(16x16,f32)  -> v_wmma_f32_16x16x4_f32
  return __builtin_amdgcn_wmma_f32_16x16x4_f32(false, a, false, b, (short)0, c,
                                               false, false);
}

__device__ __forceinline__ float gelu_f(float x) {
  return 0.5f * x * (1.0f + erff(x * 0.70710678118654752f));
}

// cosine-alphas lookup, computed analytically (ratio form cancels ac[0]).
__device__ __forceinline__ float alpha_cos(float t) {
  int ti = (int)(t * 1000.0f);
  const float s = 0.008f;
  float f0 = cosf(0.5f * 3.14159265358979f * (((float)ti / 1001.0f) + s) / (1.0f + s));
  float f1 = cosf(0.5f * 3.14159265358979f * (((float)(ti + 1) / 1001.0f) + s) / (1.0f + s));
  float ratio = (f1 * f1) / (f0 * f0);
  float beta = 1.0f - ratio;
  beta = fminf(fmaxf(beta, 0.0f), 0.9999f);
  return 1.0f - beta;
}

// ------------------------- generic strided WMMA GEMM ------------------------
// flags: 1=add bias, 2=gelu, 4=row mask (node_mask[row]), 8=edge mask
// Batch base = ptr + (z/zdiv)*s?z1 + (z%zdiv)*s?z2.
// AK1/BK1: the respective K-stride is 1 -> use 64-bit fragment loads.
template <int AK1, int BK1>
__global__ __launch_bounds__(256) void gemm_wmma(
    const float* __restrict__ A, const float* __restrict__ B,
    float* __restrict__ C, const float* __restrict__ bias,
    const unsigned char* __restrict__ mask, int M, int N, int K, long ldAm,
    long ldAk, long ldBk, long ldBn, long ldC, long sAz1, long sAz2, long sBz1,
    long sBz2, long sCz1, long sCz2, int zdiv, float alpha, int flags) {
  int z = blockIdx.z;
  int z1 = z / zdiv, z2 = z - z1 * zdiv;
  const float* Ab = A + (long)z1 * sAz1 + (long)z2 * sAz2;
  const float* Bb = B + (long)z1 * sBz1 + (long)z2 * sBz2;
  float* Cb = C + (long)z1 * sCz1 + (long)z2 * sCz2;
  int w = threadIdx.x >> 5;
  int mi = w >> 2, ni = w & 3;
  int m0 = blockIdx.y * 32 + mi * 16;
  int n0 = blockIdx.x * 64 + ni * 16;
  if (m0 >= M || n0 >= N) return;  // wave-uniform, EXEC stays all-ones
  int lane = threadIdx.x & 31;
  int sub = lane >> 4, l16 = lane & 15;
  v8f acc = {0.f, 0.f, 0.f, 0.f, 0.f, 0.f, 0.f, 0.f};
  const float* Ap = Ab + (long)(m0 + l16) * ldAm;
  const float* Bp = Bb + (long)(n0 + l16) * ldBn;
  for (int k0 = 0; k0 < K; k0 += 4) {
    int ka = k0 + 2 * sub;
    v2f a, b;
    if (AK1) {
      a = *(const v2f*)(Ap + ka);
    } else {
      a.x = Ap[(long)ka * ldAk];
      a.y = Ap[(long)(ka + 1) * ldAk];
    }
    if (BK1) {
      b = *(const v2f*)(Bp + ka);
    } else {
      b.x = Bp[(long)ka * ldBk];
      b.y = Bp[(long)(ka + 1) * ldBk];
    }
    acc = wmma4(a, b, acc);
  }
#pragma unroll
  for (int r = 0; r < 8; r++) {
    int row = m0 + r + 8 * sub;
    int col = n0 + l16;
    float v = acc[r] * alpha;
    if (flags & 1) v += bias[col];
    if (flags & 2) v = gelu_f(v);
    if (flags & 4) {
      if (!mask[row]) v = 0.f;
    }
    if (flags & 8) {
      int bb = row >> 18, ii = (row >> 9) & 511, jj = row & 511;
      if (!(mask[bb * NN + ii] && mask[bb * NN + jj])) v = 0.f;
    }
    Cb[(long)row * ldC + col] = v;
  }
}

// --------------------- fused E-FFN: 64 ->(gelu) 128 -> 64 -------------------
// One wave = 16 E-rows. A-tile (4KB, contiguous) is staged into LDS with the
// gfx1250 async global->LDS path; hidden tile lives in LDS (never hits HBM).
__global__ __launch_bounds__(256) void effn_fused(
    const float* __restrict__ Ein, const float* __restrict__ W1,
    const float* __restrict__ b1, const float* __restrict__ W2,
    const float* __restrict__ b2, float* __restrict__ Eout) {
  __shared__ float hid[8][16 * 128];    // 64 KB
  __shared__ float atile[8][16 * 64];   // 32 KB
  int w = threadIdx.x >> 5, lane = threadIdx.x & 31;
  int sub = lane >> 4, l16 = lane & 15;
  long r0 = ((long)blockIdx.x * 8 + w) * 16;
  const float* gsrc = Ein + r0 * 64;  // 16 rows x 64 f32, contiguous 4KB
#if USE_ASYNC_LDS
#pragma unroll
  for (int i = 0; i < 8; i++) {
    const float* g = gsrc + i * 128 + lane * 4;
    float* l = &atile[w][i * 128 + lane * 4];
    __builtin_amdgcn_global_load_async_to_lds_b128(
        (__attribute__((address_space(1))) vi4*)(void*)const_cast<float*>(g),
        (__attribute__((address_space(3))) vi4*)(void*)l, 0, 0);
  }
  __builtin_amdgcn_s_wait_asynccnt(0);
#else
#pragma unroll
  for (int i = 0; i < 8; i++) {
    *(float4*)&atile[w][i * 128 + lane * 4] =
        *(const float4*)(gsrc + i * 128 + lane * 4);
  }
#endif
  float areg[32];
#pragma unroll
  for (int ks = 0; ks < 16; ks++) {
    int kk = 4 * ks + 2 * sub;
    areg[2 * ks] = atile[w][l16 * 64 + kk];
    areg[2 * ks + 1] = atile[w][l16 * 64 + kk + 1];
  }
  for (int nt = 0; nt < 8; nt++) {
    v8f acc = {0.f, 0.f, 0.f, 0.f, 0.f, 0.f, 0.f, 0.f};
#pragma unroll
    for (int ks = 0; ks < 16; ks++) {
      int kk = 4 * ks + 2 * sub;
      v2f a, b;
      a.x = areg[2 * ks];
      a.y = areg[2 * ks + 1];
      b.x = W1[(long)kk * 128 + nt * 16 + l16];
      b.y = W1[(long)(kk + 1) * 128 + nt * 16 + l16];
      acc = wmma4(a, b, acc);
    }
#pragma unroll
    for (int r = 0; r < 8; r++) {
      int rl = r + 8 * sub, col = nt * 16 + l16;
      hid[w][rl * 128 + col] = gelu_f(acc[r] + b1[col]);
    }
  }
  __syncthreads();
  for (int nt = 0; nt < 4; nt++) {
    v8f acc = {0.f, 0.f, 0.f, 0.f, 0.f, 0.f, 0.f, 0.f};
#pragma unroll 4
    for (int ks = 0; ks < 32; ks++) {
      int kk = 4 * ks + 2 * sub;
      v2f a, b;
      a.x = hid[w][l16 * 128 + kk];
      a.y = hid[w][l16 * 128 + kk + 1];
      b.x = W2[(long)kk * 64 + nt * 16 + l16];
      b.y = W2[(long)(kk + 1) * 64 + nt * 16 + l16];
      acc = wmma4(a, b, acc);
    }
#pragma unroll
    for (int r = 0; r < 8; r++) {
      int rl = r + 8 * sub, col = nt * 16 + l16;
      Eout[(r0 + rl) * 64 + col] = acc[r] + b2[col];
    }
  }
}

// ------------------------------ elementwise ---------------------------------
__global__ void edge_combine(float* __restrict__ Y, const float* __restrict__ EP,
                             const float* __restrict__ BG,
                             float* __restrict__ E8,
                             const float* __restrict__ baseg,
                             const unsigned char* __restrict__ mask,
                             const float* __restrict__ yea,
                             const float* __restrict__ yem,
                             const float* __restrict__ tini) {
  long t = (long)blockIdx.x * blockDim.x + threadIdx.x;
  if (t >= (long)BSZ * NN * NN * NHH) return;
  int h = (int)(t & 7);
  long row = t >> 3;
  int jj = (int)(row & 511), ii = (int)((row >> 9) & 511), bb = (int)(row >> 18);
  long yidx = (((long)(bb * NHH + h) * NN) + ii) * NN + jj;
  float y0 = Y[yidx];
  float e1 = EP[row * 16 + h], e2 = EP[row * 16 + 8 + h];
  float Ym = y0 * (e1 + 1.0f) + e2;
  float ne = yea[bb * NHH + h] + (yem[bb * NHH + h] + 1.0f) * Ym;
  if (baseg[row * 64] != 0.0f) {
    float g1 = BG[row * 16 + h], g2 = BG[row * 16 + 8 + h];
    float tc = alpha_cos(tini[bb]);
    ne = tc * ne * g1 + (1.0f - tc) * ne + tc * g2;
  }
  E8[row * 8 + h] = ne;
  Y[yidx] = mask[bb * NN + jj] ? Ym : -__builtin_inff();
}

__global__ void softmax512(float* __restrict__ Y) {
  __shared__ float sm[256];
  long row = blockIdx.x;
  int tid = threadIdx.x;
  float* p = Y + row * 512;
  float v0 = p[tid], v1 = p[tid + 256];
  sm[tid] = fmaxf(v0, v1);
  __syncthreads();
  for (int s = 128; s > 0; s >>= 1) {
    if (tid < s) sm[tid] = fmaxf(sm[tid], sm[tid + s]);
    __syncthreads();
  }
  float mx = sm[0];
  __syncthreads();
  float e0 = expf(v0 - mx), e1 = expf(v1 - mx);
  sm[tid] = e0 + e1;
  __syncthreads();
  for (int s = 128; s > 0; s >>= 1) {
    if (tid < s) sm[tid] += sm[tid + s];
    __syncthreads();
  }
  float inv = 1.0f / sm[0];
  p[tid] = e0 * inv;
  p[tid + 256] = e1 * inv;
}

__global__ void newx_mod(const float* __restrict__ wv,
                         const float* __restrict__ yxa,
                         const float* __restrict__ yxm, float* __restrict__ o) {
  long t = (long)blockIdx.x * blockDim.x + threadIdx.x;
  if (t >= (long)BSZ * NN * DXX) return;
  int bi = (int)(t / (NN * DXX)), d = (int)(t & (DXX - 1));
  o[t] = yxa[bi * DXX + d] + (yxm[bi * DXX + d] + 1.0f) * wv[t];
}

__global__ void pack_pair(const float* __restrict__ wA, const float* __restrict__ bA,
                          const float* __restrict__ wB, const float* __restrict__ bB,
                          float* __restrict__ Wo, float* __restrict__ bo) {
  int t = blockIdx.x * blockDim.x + threadIdx.x;
  if (t < 64 * 16) {
    int k = t >> 4, j = t & 15;
    Wo[t] = (j < 8) ? wA[k * 8 + j] : wB[k * 8 + (j - 8)];
  } else if (t < 64 * 16 + 16) {
    int j = t - 64 * 16;
    bo[j] = (j < 8) ? bA[j] : bB[j - 8];
  }
}

__global__ void small_lin(const float* __restrict__ x, const float* __restrict__ W,
                          const float* __restrict__ b, float* __restrict__ o,
                          int Br, int Din, int Dout, int dogelu) {
  int t = blockIdx.x * blockDim.x + threadIdx.x;
  if (t >= Br * Dout) return;
  int bi = t / Dout, j = t - bi * Dout;
  float s = b[j];
  for (int i = 0; i < Din; i++) s += x[bi * Din + i] * W[(long)i * Dout + j];
  o[t] = dogelu ? gelu_f(s) : s;
}

__global__ void ny_combine(const float* __restrict__ y, const float* __restrict__ xz,
                           const float* __restrict__ ez, const float* __restrict__ Wyy,
                           const float* __restrict__ byy, const float* __restrict__ Wxy,
                           const float* __restrict__ bxy, const float* __restrict__ Wey,
                           const float* __restrict__ bey, float* __restrict__ o) {
  int t = blockIdx.x * blockDim.x + threadIdx.x;
  if (t >= 256) return;
  int bi = t >> 7, j = t & 127;
  float s = byy[j] + bxy[j] + bey[j];
  for (int i = 0; i < 128; i++) s += y[bi * 128 + i] * Wyy[i * 128 + j];
  for (int i = 0; i < 1024; i++) s += xz[bi * 1024 + i] * Wxy[i * 128 + j];
  for (int i = 0; i < 256; i++) s += ez[bi * 256 + i] * Wey[i * 128 + j];
  o[t] = s;
}

// out = LayerNorm(a + res) * g + b    (blockDim == D, one row per block)
__global__ void ln_res(const float* __restrict__ a, const float* __restrict__ res,
                       const float* __restrict__ g, const float* __restrict__ be,
                       float* __restrict__ o, int D) {
  __shared__ float s0[256], s1[256];
  int tid = threadIdx.x;
  long row = blockIdx.x;
  float v = a[row * D + tid] + res[row * D + tid];
  s0[tid] = v;
  s1[tid] = v * v;
  __syncthreads();
  for (int s = D >> 1; s > 0; s >>= 1) {
    if (tid < s) {
      s0[tid] += s0[tid + s];
      s1[tid] += s1[tid + s];
    }
    __syncthreads();
  }
  float mean = s0[0] / D;
  float var = s1[0] / D - mean * mean;
  o[row * D + tid] = (v - mean) * rsqrtf(var + 1e-5f) * g[tid] + be[tid];
}

__global__ void reduce_x(const float* __restrict__ X, float* __restrict__ xz) {
  __shared__ float ss[256], sq[256], mn[256], mx[256];
  int blk = blockIdx.x, b = blk >> 8, d = blk & 255, tid = threadIdx.x;
  float s = 0.f, q = 0.f, lo = __builtin_inff(), hi = -__builtin_inff();
  for (int n = tid; n < NN; n += 256) {
    float v = X[(long)b * NN * DXX + (long)n * DXX + d];
    s += v; q += v * v; lo = fminf(lo, v); hi = fmaxf(hi, v);
  }
  ss[tid] = s; sq[tid] = q; mn[tid] = lo; mx[tid] = hi;
  __syncthreads();
  for (int st = 128; st > 0; st >>= 1) {
    if (tid < st) {
      ss[tid] += ss[tid + st]; sq[tid] += sq[tid + st];
      mn[tid] = fminf(mn[tid], mn[tid + st]); mx[tid] = fmaxf(mx[tid], mx[tid + st]);
    }
    __syncthreads();
  }
  if (tid == 0) {
    float mean = ss[0] / NN;
    float var = (sq[0] - NN * mean * mean) / (NN - 1);
    xz[b * 1024 + d] = mean;
    xz[b * 1024 + 256 + d] = mn[0];
    xz[b * 1024 + 512 + d] = mx[0];
    xz[b * 1024 + 768 + d] = sqrtf(fmaxf(var, 0.f));
  }
}

__global__ void reduce_e(const float* __restrict__ E, float* __restrict__ ez) {
  __shared__ float ss[256], sq[256], mn[256], mx[256];
  int blk = blockIdx.x, b = blk >> 6, d = blk & 63, tid = threadIdx.x;
  const long CNT = (long)NN * NN;
  float s = 0.f, q = 0.f, lo = __builtin_inff(), hi = -__builtin_inff();
  for (long n = tid; n < CNT; n += 256) {
    float v = E[(long)b * CNT * DEE + n * DEE + d];
    s += v; q += v * v; lo = fminf(lo, v); hi = fmaxf(hi, v);
  }
  ss[tid] = s; sq[tid] = q; mn[tid] = lo; mx[tid] = hi;
  __syncthreads();
  for (int st = 128; st > 0; st >>= 1) {
    if (tid < st) {
      ss[tid] += ss[tid + st]; sq[tid] += sq[tid + st];
      mn[tid] = fminf(mn[tid], mn[tid + st]); mx[tid] = fmaxf(mx[tid], mx[tid + st]);
    }
    __syncthreads();
  }
  if (tid == 0) {
    float mean = ss[0] / (float)CNT;
    float var = (sq[0] - (float)CNT * mean * mean) / (float)(CNT - 1);
    ez[b * 256 + d] = mean;
    ez[b * 256 + 64 + d] = mn[0];
    ez[b * 256 + 128 + d] = mx[0];
    ez[b * 256 + 192 + d] = sqrtf(fmaxf(var, 0.f));
  }
}

// ------------------------------- host side ----------------------------------
static void gemm(hipStream_t st, const float* A, const float* B, float* C,
                 const float* bias, const unsigned char* mask, int M, int N,
                 int K, long ldAm, long ldAk, long ldBk, long ldBn, long ldC,
                 int Z, long sAz1, long sAz2, long sBz1, long sBz2, long sCz1,
                 long sCz2, int zdiv, float alpha, int flags) {
  dim3 g((N + 63) / 64, (M + 31) / 32, Z);
  bool ak1 = (ldAk == 1), bk1 = (ldBk == 1);
  if (ak1 && bk1)
    gemm_wmma<1, 1><<<g, 256, 0, st>>>(A, B, C, bias, mask, M, N, K, ldAm, ldAk,
                                       ldBk, ldBn, ldC, sAz1, sAz2, sBz1, sBz2,
                                       sCz1, sCz2, zdiv, alpha, flags);
  else if (ak1)
    gemm_wmma<1, 0><<<g, 256, 0, st>>>(A, B, C, bias, mask, M, N, K, ldAm, ldAk,
                                       ldBk, ldBn, ldC, sAz1, sAz2, sBz1, sBz2,
                                       sCz1, sCz2, zdiv, alpha, flags);
  else if (bk1)
    gemm_wmma<0, 1><<<g, 256, 0, st>>>(A, B, C, bias, mask, M, N, K, ldAm, ldAk,
                                       ldBk, ldBn, ldC, sAz1, sAz2, sBz1, sBz2,
                                       sCz1, sCz2, zdiv, alpha, flags);
  else
    gemm_wmma<0, 0><<<g, 256, 0, st>>>(A, B, C, bias, mask, M, N, K, ldAm, ldAk,
                                       ldBk, ldBn, ldC, sAz1, sAz2, sBz1, sBz2,
                                       sCz1, sCz2, zdiv, alpha, flags);
}

// workspace offsets (floats)
static const long F_Q = 0;
static const long F_K = F_Q + 262144;
static const long F_V = F_K + 262144;
static const long F_WV = F_V + 262144;
static const long F_NEWX = F_WV + 262144;
static const long F_NX = F_NEWX + 262144;
static const long F_Y = F_NX + 262144;             // 4194304
static const long F_E8 = F_Y + 4194304;            // 4194304
static const long F_NE = F_E8 + 4194304;           // 33554432
static const long F_EA = F_NE + 33554432;          // 33554432
static const long F_EB = F_EA + 33554432;          // 33554432
static const long F_EC = F_EB + 33554432;          // 33554432
static const long F_EP = F_EA;                     // alias (dead before EA use)
static const long F_BG = F_EA + 8388608;           // alias
static const long F_XA = F_EC + 33554432;          // 262144
static const long F_XH = F_XA + 262144;            // 1048576
static const long F_XF = F_XH + 1048576;           // 262144
static const long F_SM = F_XF + 262144;            // small pool
static const long S_WEM = F_SM + 0, S_BEM = F_SM + 1024;
static const long S_WBG = F_SM + 1040, S_BBG = F_SM + 2064;
static const long S_YEA = F_SM + 2080, S_YEM = F_SM + 2096;
static const long S_YXA = F_SM + 2112, S_YXM = F_SM + 2624;
static const long S_XZ = F_SM + 3136, S_EZ = F_SM + 5184;
static const long S_NY = F_SM + 5696, S_YT = F_SM + 5952, S_NY2 = F_SM + 6208;
static const long S_YA = F_SM + 6464, S_YH = F_SM + 6720, S_YF = F_SM + 7744;
static const long S_Y1 = F_SM + 8000;

extern "C" void kernel_launch(void* const* d_in, const int* in_sizes, int n_in,
                              void* d_out, int out_size, void* d_ws,
                              size_t ws_size, hipStream_t stream) {
  (void)in_sizes; (void)n_in; (void)out_size; (void)ws_size;
  const float* X0 = (const float*)d_in[0];
  const float* X1 = (const float*)d_in[1];
  const float* E = (const float*)d_in[2];
  const float* yin = (const float*)d_in[3];
  const float* BGin = (const float*)d_in[4];
  const float* tini = (const float*)d_in[5];
  const unsigned char* msk = (const unsigned char*)d_in[6];
  auto PF = [&](int leaf) -> const float* { return (const float*)d_in[leaf]; };
  // attn leaves: b = 7+2i, w = 7+2i+1
  auto AB = [&](int i) { return PF(7 + 2 * i); };
  auto AW = [&](int i) { return PF(7 + 2 * i + 1); };
  enum { A_BGA = 0, A_BGM, A_EADD, A_EMUL, A_EOUT, A_EY, A_K, A_Q, A_V, A_XOUT,
         A_XY, A_YEA, A_YEM, A_YO1, A_YO2, A_YXA, A_YXM, A_YY };
  auto LB = [&](int tl, int i) { return PF((tl ? 67 : 43) + 2 * i); };
  auto LW = [&](int tl, int i) { return PF((tl ? 67 : 43) + 2 * i + 1); };
  auto NB = [&](int tl, int i) { return PF((tl ? 67 : 43) + 12 + 2 * i); };
  auto NG = [&](int tl, int i) { return PF((tl ? 67 : 43) + 12 + 2 * i + 1); };
  enum { L_E1 = 0, L_E2, L_X1, L_X2, L_Y1, L_Y2 };  // lin order
  enum { N_E1 = 0, N_E2, N_X1, N_X2, N_Y1, N_Y2 };  // norm order
  float* ws = (float*)d_ws;
  float* outp = (float*)d_out;
  const long EDGES = (long)BSZ * NN * NN;  // 524288

  // --- y projections needed early ---
  small_lin<<<1, 128, 0, stream>>>(yin, AW(A_YEA), AB(A_YEA), ws + S_YEA, 2, 128, 8, 0);
  small_lin<<<1, 128, 0, stream>>>(yin, AW(A_YEM), AB(A_YEM), ws + S_YEM, 2, 128, 8, 0);
  small_lin<<<4, 128, 0, stream>>>(yin, AW(A_YXA), AB(A_YXA), ws + S_YXA, 2, 128, 256, 0);
  small_lin<<<4, 128, 0, stream>>>(yin, AW(A_YXM), AB(A_YXM), ws + S_YXM, 2, 128, 256, 0);
  // --- pack 64->8 projections into 64->16 GEMMs ---
  pack_pair<<<9, 128, 0, stream>>>(AW(A_EMUL), AB(A_EMUL), AW(A_EADD), AB(A_EADD), ws + S_WEM, ws + S_BEM);
  pack_pair<<<9, 128, 0, stream>>>(AW(A_BGM), AB(A_BGM), AW(A_BGA), AB(A_BGA), ws + S_WBG, ws + S_BBG);
  // --- Q/K/V projections (Q uses X0, K/V use X1), masked ---
  gemm(stream, X0, AW(A_Q), ws + F_Q, AB(A_Q), msk, 1024, 256, 256, 256, 1, 256, 1, 256, 1, 0, 0, 0, 0, 0, 0, 1, 1.0f, 1 | 4);
  gemm(stream, X1, AW(A_K), ws + F_K, AB(A_K), msk, 1024, 256, 256, 256, 1, 256, 1, 256, 1, 0, 0, 0, 0, 0, 0, 1, 1.0f, 1 | 4);
  gemm(stream, X1, AW(A_V), ws + F_V, AB(A_V), msk, 1024, 256, 256, 256, 1, 256, 1, 256, 1, 0, 0, 0, 0, 0, 0, 1, 1.0f, 1 | 4);
  // --- Y = Q K^T / sqrt(df), batched over (b,h): zdiv=8 ---
  gemm(stream, ws + F_Q, ws + F_K, ws + F_Y, nullptr, nullptr, 512, 512, 32,
       256, 1, 1, 256, 512, 16, 262144, 32, 262144, 32, 2097152, 262144, 8,
       0.17677669529663688f, 0);
  // --- edge projections: E->16 (e_mul|e_add), base_GRN->16 (bg_mul|bg_add) ---
  gemm(stream, E, ws + S_WEM, ws + F_EP, ws + S_BEM, msk, (int)EDGES, 16, 64,
       64, 1, 16, 1, 16, 1, 0, 0, 0, 0, 0, 0, 1, 1.0f, 1 | 8);
  gemm(stream, BGin, ws + S_WBG, ws + F_BG, ws + S_BBG, msk, (int)EDGES, 16, 64,
       64, 1, 16, 1, 16, 1, 0, 0, 0, 0, 0, 0, 1, 1.0f, 1 | 8);
  // --- combine: Y', newE8, key-mask ---
  edge_combine<<<(unsigned)((EDGES * 8 + 255) / 256), 256, 0, stream>>>(
      ws + F_Y, ws + F_EP, ws + F_BG, ws + F_E8, BGin, msk, ws + S_YEA, ws + S_YEM, tini);
  // --- softmax over keys ---
  softmax512<<<BSZ * NHH * NN, 256, 0, stream>>>(ws + F_Y);
  // --- wV = attn @ V, batched (b,h) ---
  gemm(stream, ws + F_Y, ws + F_V, ws + F_WV, nullptr, nullptr, 512, 32, 512,
       512, 1, 256, 1, 256, 16, 2097152, 262144, 262144, 32, 262144, 32, 8,
       1.0f, 0);
  // --- newX = y_x_add + (y_x_mul+1)*wV ; nX = x_out(newX)*xm ---
  newx_mod<<<(262144 + 255) / 256, 256, 0, stream>>>(ws + F_WV, ws + S_YXA, ws + S_YXM, ws + F_NEWX);
  gemm(stream, ws + F_NEWX, AW(A_XOUT), ws + F_NX, AB(A_XOUT), msk, 1024, 256,
       256, 256, 1, 256, 1, 256, 1, 0, 0, 0, 0, 0, 0, 1, 1.0f, 1 | 4);
  // --- nE = e_out(newE8)*em ---
  gemm(stream, ws + F_E8, AW(A_EOUT), ws + F_NE, AB(A_EOUT), msk, (int)EDGES,
       64, 8, 8, 1, 64, 1, 64, 1, 0, 0, 0, 0, 0, 0, 1, 1.0f, 1 | 8);
  // --- new_y = y_y(y) + x_y(stats(X1)) + e_y(stats(E)), then y_out MLP ---
  reduce_x<<<BSZ * 256, 256, 0, stream>>>(X1, ws + S_XZ);
  reduce_e<<<BSZ * 64, 256, 0, stream>>>(E, ws + S_EZ);
  ny_combine<<<2, 128, 0, stream>>>(yin, ws + S_XZ, ws + S_EZ, AW(A_YY), AB(A_YY),
                                    AW(A_XY), AB(A_XY), AW(A_EY), AB(A_EY), ws + S_NY);
  small_lin<<<2, 128, 0, stream>>>(ws + S_NY, AW(A_YO1), AB(A_YO1), ws + S_YT, 2, 128, 128, 1);
  small_lin<<<2, 128, 0, stream>>>(ws + S_YT, AW(A_YO2), AB(A_YO2), ws + S_NY2, 2, 128, 128, 0);

  // ================= transformer layers (tl1 on X0/E/y, tl2 on X1/E1/y1) ====
  const float* Xin[2] = {X0, X1};
  float* Xout[2] = {outp + 0, outp + 262144};
  const float* Ein[2] = {E, ws + F_EC};
  float* Eout[2] = {ws + F_EC, outp + 524288};
  const float* Yin2[2] = {yin, ws + S_Y1};
  float* Yout2[2] = {ws + S_Y1, outp + 34078720};
  for (int tl = 0; tl < 2; tl++) {
    // X branch
    ln_res<<<1024, 256, 0, stream>>>(Xin[tl], ws + F_NX, NG(tl, N_X1), NB(tl, N_X1), ws + F_XA, 256);
    gemm(stream, ws + F_XA, LW(tl, L_X1), ws + F_XH, LB(tl, L_X1), nullptr, 1024,
         1024, 256, 256, 1, 1024, 1, 1024, 1, 0, 0, 0, 0, 0, 0, 1, 1.0f, 1 | 2);
    gemm(stream, ws + F_XH, LW(tl, L_X2), ws + F_XF, LB(tl, L_X2), nullptr, 1024,
         256, 1024, 1024, 1, 256, 1, 256, 1, 0, 0, 0, 0, 0, 0, 1, 1.0f, 1);
    ln_res<<<1024, 256, 0, stream>>>(ws + F_XA, ws + F_XF, NG(tl, N_X2), NB(tl, N_X2), Xout[tl], 256);
    // E branch (fused FFN keeps 268MB hidden in LDS)
    ln_res<<<(unsigned)EDGES, 64, 0, stream>>>(Ein[tl], ws + F_NE, NG(tl, N_E1), NB(tl, N_E1), ws + F_EA, 64);
    effn_fused<<<(unsigned)(EDGES / 128), 256, 0, stream>>>(ws + F_EA, LW(tl, L_E1), LB(tl, L_E1), LW(tl, L_E2), LB(tl, L_E2), ws + F_EB);
    ln_res<<<(unsigned)EDGES, 64, 0, stream>>>(ws + F_EA, ws + F_EB, NG(tl, N_E2), NB(tl, N_E2), Eout[tl], 64);
    // y branch
    ln_res<<<2, 128, 0, stream>>>(Yin2[tl], ws + S_NY2, NG(tl, N_Y1), NB(tl, N_Y1), ws + S_YA, 128);
    small_lin<<<8, 128, 0, stream>>>(ws + S_YA, LW(tl, L_Y1), LB(tl, L_Y1), ws + S_YH, 2, 128, 512, 1);
    small_lin<<<2, 128, 0, stream>>>(ws + S_YH, LW(tl, L_Y2), LB(tl, L_Y2), ws + S_YF, 2, 512, 128, 0);
    ln_res<<<2, 128, 0, stream>>>(ws + S_YA, ws + S_YF, NG(tl, N_Y2), NB(tl, N_Y2), Yout2[tl], 128);
  }
}